// GCU_58110907515157
// MI455X (gfx1250) — compile-verified
//
#include <hip/hip_runtime.h>
#include <math.h>
#include <stdint.h>

#define EPS 1e-7f
// Shapes (fixed by reference): B=4, H=W=64 -> HW=4096, d=V=O=64
#define NB 4
#define HW 4096
#define DV 64

typedef __attribute__((ext_vector_type(2))) float v2f;
typedef __attribute__((ext_vector_type(8))) float v8f;

static __device__ __forceinline__ v8f wmma_f32_4(v2f a, v2f b, v8f c) {
  // V_WMMA_F32_16X16X4_F32 : D = A(16x4) x B(4x16) + C(16x16), all f32
  return __builtin_amdgcn_wmma_f32_16x16x4_f32(
      /*neg_a=*/false, a, /*neg_b=*/false, b,
      /*c_mod=*/(short)0, c, /*reuse_a=*/false, /*reuse_b=*/false);
}

// ---- Kernel 1: per-(i,v) precompute: A=1/var, WA=W/var, and per-v sums
// S2[v]=sum_i A^2, S1[v]=sum_i W*A^2, S0[v]=sum_i W^2*A^2
__global__ void k1_precomp(const float* __restrict__ Wm, const float* __restrict__ var,
                           float* __restrict__ Abuf, float* __restrict__ WAbuf,
                           float* __restrict__ S2, float* __restrict__ S1,
                           float* __restrict__ S0) {
  int v = threadIdx.x;  // 64 threads
  float s2 = 0.f, s1 = 0.f, s0 = 0.f;
  for (int i = 0; i < DV; ++i) {
    float w = Wm[i * DV + v];
    float a = 1.0f / var[i * DV + v];
    Abuf[i * DV + v] = a;
    WAbuf[i * DV + v] = w * a;
    float a2 = a * a;
    s2 += a2;
    s1 += w * a2;
    s0 += w * w * a2;
  }
  S2[v] = s2; S1[v] = s1; S0[v] = s0;
}

// ---- Kernel 2: responsibilities Q[b,n,v] = softmax_v(-0.5*(q - min_v q)),
// q = x^2*S2 - 2x*S1 + S0 with x = X[b,n,v]. One wave32 per row (2 v per lane).
__global__ void k2_resp(const float* __restrict__ X, const float* __restrict__ S2,
                        const float* __restrict__ S1, const float* __restrict__ S0,
                        float* __restrict__ Q) {
  int gw = (blockIdx.x * blockDim.x + threadIdx.x) >> 5;  // global wave = row id
  int lane = threadIdx.x & 31;
  int b = gw >> 12;          // / 4096
  int n = gw & (HW - 1);
  const float* xr = X + ((size_t)(b * HW + n)) * DV;
  int v0 = lane, v1 = lane + 32;
  float x0 = xr[v0], x1 = xr[v1];
  float q0 = x0 * x0 * S2[v0] - 2.f * x0 * S1[v0] + S0[v0];
  float q1 = x1 * x1 * S2[v1] - 2.f * x1 * S1[v1] + S0[v1];
  float m = fminf(q0, q1);
  for (int off = 16; off >= 1; off >>= 1) m = fminf(m, __shfl_xor(m, off, 32));
  float e0 = __expf(-0.5f * (q0 - m));
  float e1 = __expf(-0.5f * (q1 - m));
  float s = e0 + e1;
  for (int off = 16; off >= 1; off >>= 1) s += __shfl_xor(s, off, 32);
  float inv = 1.0f / s;
  float* qr = Q + ((size_t)(b * HW + n)) * DV;
  qr[v0] = e0 * inv;
  qr[v1] = e1 * inv;
}

// ---- Kernel 3: R[b,m1,v] = sum_m0 A[m0,v]*Q[b,m1*64+m0,v];
// atomically: Qsum[b,v] += sum_m0 Q ; T[b,v] += sum_m0 WA[m0,v]*Q
__global__ void k3_reduce(const float* __restrict__ Q, const float* __restrict__ Abuf,
                          const float* __restrict__ WAbuf, float* __restrict__ R,
                          float* __restrict__ T, float* __restrict__ Qsum) {
  int b = blockIdx.x >> 6;
  int m1 = blockIdx.x & 63;
  int v = threadIdx.x;  // 64 threads
  float r = 0.f, p = 0.f, t = 0.f;
  const float* qbase = Q + ((size_t)(b * HW + m1 * 64)) * DV + v;
  for (int m0 = 0; m0 < 64; ++m0) {
    float qv = qbase[m0 * DV];
    r += Abuf[m0 * DV + v] * qv;
    p += qv;
    t += WAbuf[m0 * DV + v] * qv;
  }
  R[(b * 64 + m1) * DV + v] = r;
  atomicAdd(&Qsum[b * DV + v], p);
  atomicAdd(&T[b * DV + v], t);
}

// ---- Kernel 4: z[b,a,v] = sum_m1 X[b,a*64+m1,v]*R[b,m1,v] - T[b,v];
// z = (z+EPS)/(Qsum+EPS); norm[v]=sum_a z^2; Z = z/norm. One block per b.
__global__ void k4_z(const float* __restrict__ X, const float* __restrict__ R,
                     const float* __restrict__ T, const float* __restrict__ Qsum,
                     float* __restrict__ Z) {
  __shared__ float zbuf[64 * 65];  // padded to kill stride-64 bank conflicts
  __shared__ float nbuf[64];
  int b = blockIdx.x;
  int tid = threadIdx.x;  // 256
  for (int it = 0; it < 16; ++it) {
    int ent = it * 256 + tid;
    int a = ent >> 6, v = ent & 63;
    float acc = 0.f;
    const float* xb = X + ((size_t)(b * HW + a * 64)) * DV + v;
    const float* rb = R + (b * 64) * DV + v;
    for (int m1 = 0; m1 < 64; ++m1) acc += xb[m1 * DV] * rb[m1 * DV];
    float z = acc - T[b * DV + v];
    z = (z + EPS) / (Qsum[b * DV + v] + EPS);
    zbuf[a * 65 + v] = z;
  }
  __syncthreads();
  if (tid < 64) {
    float nrm = 0.f;
    for (int a = 0; a < 64; ++a) { float z = zbuf[a * 65 + tid]; nrm += z * z; }
    nbuf[tid] = nrm;
  }
  __syncthreads();
  for (int it = 0; it < 16; ++it) {
    int ent = it * 256 + tid;
    int a = ent >> 6, v = ent & 63;
    Z[((size_t)b * 64 + a) * DV + v] = zbuf[a * 65 + v] / nbuf[v];
  }
}

// ---- Kernel 5: per batch, three 64^3 f32 GEMMs via V_WMMA_F32_16X16X4_F32:
// Adj = Z^T Z, O1 = Z^T weight, Zo = relu(Adj @ O1). 16 waves, one 16x16 tile each.
__global__ void __launch_bounds__(512) k5_graph(const float* __restrict__ Z,
                                                const float* __restrict__ Wt,
                                                float* __restrict__ Zo) {
  __shared__ float Zs[4096];
  __shared__ float Ws[4096];
  __shared__ float AdjS[4096];
  __shared__ float O1S[4096];
  int b = blockIdx.x;
  int tid = threadIdx.x;
  for (int i = tid; i < 4096; i += 512) {
    Zs[i] = Z[(size_t)b * 4096 + i];
    Ws[i] = Wt[i];
  }
  __syncthreads();
  int wave = tid >> 5, lane = tid & 31;
  int half = lane >> 4, lp = lane & 15;
  int mt = wave >> 2, nt = wave & 3;
  int mrow = mt * 16 + lp;   // A-matrix row (M) for this lane
  int ncol = nt * 16 + lp;   // B/D column (N) for this lane

  // GEMM1: Adj[M=v,N=u] = sum_d Z[d,v]*Z[d,u]; GEMM2: O1[M=v,N=o] = sum_d Z[d,v]*Wt[d,o]
  v8f c1 = {}; v8f c2 = {};
  for (int kk = 0; kk < 64; kk += 4) {
    int k0 = kk + half * 2;  // A layout: lanes 0-15 carry K={k0,k0+1}, lanes 16-31 K={k0+2,k0+3}
    v2f a, b1, b2;
    a.x = Zs[k0 * 64 + mrow];        a.y = Zs[(k0 + 1) * 64 + mrow];
    b1.x = Zs[k0 * 64 + ncol];       b1.y = Zs[(k0 + 1) * 64 + ncol];
    b2.x = Ws[k0 * 64 + ncol];       b2.y = Ws[(k0 + 1) * 64 + ncol];
    c1 = wmma_f32_4(a, b1, c1);
    c2 = wmma_f32_4(a, b2, c2);
  }
  for (int r = 0; r < 8; ++r) {  // C/D layout: M = mt*16 + r + 8*half, N = ncol
    int M = mt * 16 + r + half * 8;
    AdjS[M * 64 + ncol] = c1[r];
    O1S[M * 64 + ncol] = c2[r];
  }
  __syncthreads();

  // GEMM3: out2[M=v,N=o] = sum_u Adj[v,u]*O1[u,o]; Zo = relu(out2)
  v8f c3 = {};
  for (int kk = 0; kk < 64; kk += 4) {
    int k0 = kk + half * 2;
    v2f a, bb;
    a.x = AdjS[mrow * 64 + k0];      a.y = AdjS[mrow * 64 + k0 + 1];
    bb.x = O1S[k0 * 64 + ncol];      bb.y = O1S[(k0 + 1) * 64 + ncol];
    c3 = wmma_f32_4(a, bb, c3);
  }
  for (int r = 0; r < 8; ++r) {
    int M = mt * 16 + r + half * 8;
    Zo[(size_t)b * 4096 + M * 64 + ncol] = fmaxf(c3[r], 0.f);
  }
}

// ---- Kernel 6: big GEMM Xn[b,n,o] = sum_v Q[b,n,v]*Zo[b,v,o], M=4096 N=64 K=64 per b,
// one wave per 16x16 tile; Zo[b] (16KB) staged once per block into LDS via gfx1250
// async-copy (GLOBAL_LOAD_ASYNC_TO_LDS_B128, tracked with ASYNCcnt); store fused
// with transpose to out[b,o,h,w]. All 8 waves of a block share the same batch b
// (1024 tiles per batch, 8 tiles per block -> 128 blocks per batch).
__global__ void __launch_bounds__(256) k6_out(const float* __restrict__ Q,
                                              const float* __restrict__ Zo,
                                              float* __restrict__ out) {
  __shared__ float Zs[4096];
  int tid = threadIdx.x;
  int b = blockIdx.x >> 7;  // 128 blocks per batch
  const float* zb = Zo + (size_t)b * 4096;

  // Async-stage Zo[b] (4096 floats) into LDS: 256 threads x 4 chunks of 16B.
  // LDS byte offset = low 32 bits of the flat pointer (LDS aperture maps addr[31:0]).
#pragma unroll
  for (int it = 0; it < 4; ++it) {
    int f = (it * 256 + tid) * 4;  // float index, 16B-aligned per chunk
    const float* gp = zb + f;
    unsigned lds_off = (unsigned)(uintptr_t)(Zs + f);
    asm volatile("global_load_async_to_lds_b128 %0, %1, off"
                 :
                 : "v"(lds_off), "v"(gp)
                 : "memory");
  }
  asm volatile("s_wait_asynccnt 0" ::: "memory");
  __syncthreads();

  int gw = (blockIdx.x * 256 + tid) >> 5;  // global wave / tile id
  int lane = tid & 31;
  int rem = gw & 1023;
  int mt = rem >> 2, nt = rem & 3;
  int half = lane >> 4, lp = lane & 15;
  int n0 = mt * 16 + lp;   // A-matrix row (pixel index n)
  int ncol = nt * 16 + lp; // output channel o
  const float* qb = Q + (size_t)b * HW * DV;
  __builtin_prefetch(qb + (size_t)n0 * DV, 0, 3);  // global_prefetch_b8 on Q rows
  v8f c = {};
  for (int kk = 0; kk < 64; kk += 4) {
    int k0 = kk + half * 2;
    v2f a, bb;
    const float* qr = qb + (size_t)n0 * DV + k0;
    a.x = qr[0];  a.y = qr[1];
    bb.x = Zs[k0 * 64 + ncol];  bb.y = Zs[(k0 + 1) * 64 + ncol];
    c = wmma_f32_4(a, bb, c);
  }
  for (int r = 0; r < 8; ++r) {
    int n = mt * 16 + r + half * 8;                        // pixel n = h*64+w
    out[((size_t)b * 64 + ncol) * (size_t)HW + n] = c[r];  // out[b,o,h,w]
  }
}

extern "C" void kernel_launch(void* const* d_in, const int* in_sizes, int n_in,
                              void* d_out, int out_size, void* d_ws, size_t ws_size,
                              hipStream_t stream) {
  (void)in_sizes; (void)n_in; (void)out_size; (void)ws_size;
  const float* X   = (const float*)d_in[0];  // (4,64,64,64)
  const float* Wm  = (const float*)d_in[1];  // (64,64)
  const float* var = (const float*)d_in[2];  // (64,64)
  const float* wt  = (const float*)d_in[3];  // (64,64)
  float* out = (float*)d_out;
  float* ws = (float*)d_ws;

  // Workspace layout (floats); total ~1.11M floats (~4.43 MB)
  float* Abuf  = ws;                 // 4096
  float* WAbuf = ws + 4096;          // 4096
  float* S2    = ws + 8192;          // 64
  float* S1    = ws + 8256;          // 64
  float* S0    = ws + 8320;          // 64
  float* Qb    = ws + 8384;          // 1048576
  float* Rb    = Qb + 1048576;       // 16384
  float* Tb    = Rb + 16384;         // 256
  float* Qs    = Tb + 256;           // 256
  float* Zb    = Qs + 256;           // 16384
  float* Zo    = Zb + 16384;         // 16384

  // Zero the atomic accumulators (T and Qsum are contiguous: 512 floats)
  hipMemsetAsync(Tb, 0, 512 * sizeof(float), stream);

  k1_precomp<<<1, 64, 0, stream>>>(Wm, var, Abuf, WAbuf, S2, S1, S0);
  k2_resp<<<(NB * HW) / 8, 256, 0, stream>>>(X, S2, S1, S0, Qb);
  k3_reduce<<<NB * 64, 64, 0, stream>>>(Qb, Abuf, WAbuf, Rb, Tb, Qs);
  k4_z<<<NB, 256, 0, stream>>>(X, Rb, Tb, Qs, Zb);
  k5_graph<<<NB, 512, 0, stream>>>(Zb, wt, Zo);
  k6_out<<<(NB * 1024) / 8, 256, 0, stream>>>(Qb, Zo, out);
}